// BatchGraphEncoder_72919954751665
// MI455X (gfx1250) — compile-verified
//
#include <hip/hip_runtime.h>

// ---------------------------------------------------------------------------
// Problem: B=2048, N=512, D_E=D_R=128, OUT_DIM=384.
// alpha == 1 exactly (singleton softmax segments), so:
//   out[n, 0:128]   = sum_b h[b,:]          (broadcast over n)
//   out[n, 128:256] = sum_b r[b,:]          (broadcast over n)
//   out[n, 256:384] = sum_b t[b,n,:]
// Memory-bound: 512 MB of t at 23.3 TB/s ~= 22 us floor. The B-reduction is
// fed through v_wmma_f32_16x16x4_f32 with an all-ones A matrix so the matrix
// pipe (not VALU) does the accumulation at full f32 precision.
// ---------------------------------------------------------------------------

typedef float v2f __attribute__((ext_vector_type(2)));
typedef float v8f __attribute__((ext_vector_type(8)));

#define B_ROWS 2048
#define N_COLS 512
#define D_E    128
#define OUT_DIM 384
#define T_ROW_STRIDE (N_COLS * D_E)  /* 65536 floats between consecutive b */

// Kernel A: 8-way partial column sums of h and r into workspace.
// grid = (8, 2): x = batch chunk, y = 0 -> h, 1 -> r. block = 128 (one d per thread).
__global__ void colsum_hr_kernel(const float* __restrict__ h,
                                 const float* __restrict__ r,
                                 float* __restrict__ ws) {
  const float* __restrict__ src = (blockIdx.y == 0) ? h : r;
  const int d = threadIdx.x;          // 0..127
  const int g = blockIdx.x;           // 0..7
  const int b0 = g * (B_ROWS / 8);
  float s = 0.f;
#pragma unroll 8
  for (int b = 0; b < B_ROWS / 8; ++b)
    s += src[(size_t)(b0 + b) * D_E + d];
  ws[((size_t)blockIdx.y * 8 + g) * D_E + d] = s;
}

// Kernel B: combine the 8 partials and broadcast into all 512 output rows.
// grid = 512 (one n per block), block = 128 (one d per thread).
__global__ void broadcast_hr_kernel(const float* __restrict__ ws,
                                    float* __restrict__ out) {
  const int n = blockIdx.x;
  const int d = threadIdx.x;
  float sh = 0.f, sr = 0.f;
#pragma unroll
  for (int g = 0; g < 8; ++g) {
    sh += ws[g * D_E + d];
    sr += ws[(8 + g) * D_E + d];
  }
  out[(size_t)n * OUT_DIM + d]       = sh;
  out[(size_t)n * OUT_DIM + D_E + d] = sr;
}

// Kernel C: out_t[n, d] = sum_b t[b, n, d] using v_wmma_f32_16x16x4_f32.
// A = ones(16x4)  =>  D[m, j] = sum_k B[k, j]; B column j lives in lanes j and
// j+16 (2 VGPRs each = 4 K-slots). Since A is all-ones, K placement within a
// column is irrelevant, so each lane just streams 2 strided f32 loads per
// 4 batch rows (64-B coalesced per half-wave) straight into the B fragment.
// grid = 512 (one n per block), block = 256 = 8 waves; wave w owns d = 16w..16w+15.
__global__ void __launch_bounds__(256)
reduce_t_wmma_kernel(const float* __restrict__ t, float* __restrict__ out) {
  const int n    = blockIdx.x;
  const int wave = threadIdx.x >> 5;
  const int lane = threadIdx.x & 31;
  const int j    = lane & 15;         // column (d offset) this lane feeds
  const int kh   = lane >> 4;         // which K half-slot this lane supplies
  const int d0   = wave * 16;

  const float* __restrict__ p =
      t + (size_t)n * D_E + d0 + j + (size_t)kh * T_ROW_STRIDE;

  v8f acc = {};
  const v2f ones = {1.f, 1.f};

  for (int b = 0; b < B_ROWS; b += 16) {
    // 8 independent loads per wave-iteration (4 b-rows per WMMA, 4 WMMAs).
    float x0 = p[(size_t) 0 * T_ROW_STRIDE];
    float x1 = p[(size_t) 2 * T_ROW_STRIDE];
    float x2 = p[(size_t) 4 * T_ROW_STRIDE];
    float x3 = p[(size_t) 6 * T_ROW_STRIDE];
    float x4 = p[(size_t) 8 * T_ROW_STRIDE];
    float x5 = p[(size_t)10 * T_ROW_STRIDE];
    float x6 = p[(size_t)12 * T_ROW_STRIDE];
    float x7 = p[(size_t)14 * T_ROW_STRIDE];
    p += (size_t)16 * T_ROW_STRIDE;
    if (b + 16 < B_ROWS)
      __builtin_prefetch(p, 0, 0);    // global_prefetch_b8: pull next chunk

    v2f b0 = {x0, x1};
    v2f b1 = {x2, x3};
    v2f b2 = {x4, x5};
    v2f b3 = {x6, x7};
    // 8 args: (neg_a, A, neg_b, B, c_mod, C, reuse_a, reuse_b)
    acc = __builtin_amdgcn_wmma_f32_16x16x4_f32(false, ones, false, b0,
                                                (short)0, acc, false, false);
    acc = __builtin_amdgcn_wmma_f32_16x16x4_f32(false, ones, false, b1,
                                                (short)0, acc, false, false);
    acc = __builtin_amdgcn_wmma_f32_16x16x4_f32(false, ones, false, b2,
                                                (short)0, acc, false, false);
    acc = __builtin_amdgcn_wmma_f32_16x16x4_f32(false, ones, false, b3,
                                                (short)0, acc, false, false);
  }

  // Every D element in column j equals the column sum; VGPR0 of lanes 0..15
  // holds (M=0, N=lane). One store per column.
  if (lane < 16)
    out[(size_t)n * OUT_DIM + 2 * D_E + d0 + lane] = acc[0];
}

extern "C" void kernel_launch(void* const* d_in, const int* in_sizes, int n_in,
                              void* d_out, int out_size, void* d_ws, size_t ws_size,
                              hipStream_t stream) {
  const float* h = (const float*)d_in[0];   // [2048, 128]
  const float* r = (const float*)d_in[1];   // [2048, 128]
  const float* t = (const float*)d_in[2];   // [2048, 512, 128]
  // d_in[3..5] = w_i, w_j, w_k: mathematically dead (alpha == 1 exactly).
  float* out = (float*)d_out;               // [512, 384]
  float* ws  = (float*)d_ws;                // 16*128 floats = 8 KB used

  colsum_hr_kernel   <<<dim3(8, 2), 128, 0, stream>>>(h, r, ws);
  broadcast_hr_kernel<<<N_COLS,     128, 0, stream>>>(ws, out);
  reduce_t_wmma_kernel<<<N_COLS,    256, 0, stream>>>(t, out);
}